// Steerable_Encoder_18820546691859
// MI455X (gfx1250) — compile-verified
//
#include <hip/hip_runtime.h>

typedef __attribute__((ext_vector_type(2))) float v2f;
typedef __attribute__((ext_vector_type(8))) float v8f;

#define N_CTX  8192
#define N_GRID 16384

// Each wave: one 16-row grid tile (M), loops over 512 context tiles (N=16).
// d2 tile via V_WMMA_F32_16X16X4_F32 using the augmented-coordinate trick,
// with the exp2 scale folded into the (loop-invariant) A matrix:
//   C = -0.5*log2(e)
//   A[m,:] = C * [-2gx, -2gy, gx^2+gy^2, 1]     (16x4)
//   B[:,n] =     [ xi ,  yi , 1, xi^2+yi^2 ]    (4x16)
//   D[m,n] = C * ||grid_m - X_n||^2             (exp2 argument, no extra VALU)
// Then e = v_exp_f32(D) on the 8 D VGPRs, accumulate 3 channels with dual FMAs.
__global__ __launch_bounds__(256) void steerable_encoder_kernel(
    const float* __restrict__ X, const float* __restrict__ Y,
    float* __restrict__ out)
{
    const int lane = threadIdx.x & 31;
    const int wave = threadIdx.x >> 5;
    const int m    = lane & 15;          // row (lanes 0-15) / col index (lane%16)
    const bool hi  = lane >= 16;

    const int mBase = (blockIdx.x * 8 + wave) * 16;   // first grid row of this tile
    const int g = mBase + m;                          // grid point index for row m
    const int j = g & 127;                            // x index
    const int k = g >> 7;                             // y index
    const float step = 4.0f / 127.0f;
    const float gx = -2.0f + step * (float)j;
    const float gy =  2.0f - step * (float)k;
    const float g2 = gx * gx + gy * gy;

    const float C = -0.72134752044448170368f;  // -0.5 * log2(e)

    // A matrix (pre-scaled by C), 16x4 f32: lanes 0-15 -> K=0,1 ; lanes 16-31 -> K=2,3
    v2f a;
    a.x = hi ? (C * g2) : (-2.0f * C * gx);
    a.y = hi ? C        : (-2.0f * C * gy);

    float acc0[8], acc1[8], acc2[8];
#pragma unroll
    for (int r = 0; r < 8; ++r) { acc0[r] = 0.0f; acc1[r] = 0.0f; acc2[r] = 0.0f; }

    const float2* __restrict__ X2 = (const float2*)X;
    const float2* __restrict__ Y2 = (const float2*)Y;

#pragma unroll 2
    for (int n0 = 0; n0 < N_CTX; n0 += 16) {
        const int ci = n0 + m;           // context point for this lane's column
        float2 xv = X2[ci];
        float2 yv = Y2[ci];
        float x2 = xv.x * xv.x + xv.y * xv.y;

        // B matrix, 4x16 f32: lanes 0-15 -> K=0,1 ; lanes 16-31 -> K=2,3
        v2f b;
        b.x = hi ? 1.0f : xv.x;
        b.y = hi ? x2   : xv.y;

        v8f c = {0.f, 0.f, 0.f, 0.f, 0.f, 0.f, 0.f, 0.f};
        // (neg_a, A, neg_b, B, c_mod, C, reuse_a, reuse_b)
        v8f d = __builtin_amdgcn_wmma_f32_16x16x4_f32(
            false, a, false, b, (short)0, c, false, false);

        // D layout: VGPR r of lane l holds D[r + 8*(l>=16)][l%16] = C*d2
#pragma unroll
        for (int r = 0; r < 8; ++r) {
            float e = __builtin_amdgcn_exp2f(d[r]);       // exp(-0.5*d2)
            acc0[r] += e;                                  // density channel
            acc1[r] = fmaf(e, yv.x, acc1[r]);              // Gram @ Y[:,0]
            acc2[r] = fmaf(e, yv.y, acc2[r]);              // Gram @ Y[:,1]
        }
    }

    // Reduce over the 16 columns held across lanes of each half-wave.
    // xor masks 8,4,2,1 stay within lanes [0,15] and [16,31].
#pragma unroll
    for (int r = 0; r < 8; ++r) {
        float s0 = acc0[r], s1 = acc1[r], s2 = acc2[r];
#pragma unroll
        for (int off = 8; off >= 1; off >>= 1) {
            s0 += __shfl_xor(s0, off, 32);
            s1 += __shfl_xor(s1, off, 32);
            s2 += __shfl_xor(s2, off, 32);
        }
        if (m == r) {
            const int row = r + (hi ? 8 : 0);
            const int gg  = mBase + row;
            out[gg]              = s0;
            out[N_GRID + gg]     = s1 / s0;   // normalize by density
            out[2 * N_GRID + gg] = s2 / s0;
        }
    }
}

extern "C" void kernel_launch(void* const* d_in, const int* in_sizes, int n_in,
                              void* d_out, int out_size, void* d_ws, size_t ws_size,
                              hipStream_t stream) {
    const float* X = (const float*)d_in[0];   // (8192, 2) fp32
    const float* Y = (const float*)d_in[1];   // (8192, 2) fp32
    float* out = (float*)d_out;               // (1, 3, 128, 128) fp32 = 49152

    // 1024 M-tiles of 16 grid rows; 8 waves (256 threads) per block -> 128 blocks
    steerable_encoder_kernel<<<dim3(128), dim3(256), 0, stream>>>(X, Y, out);
}